// SelfAttention_53927609368795
// MI455X (gfx1250) — compile-verified
//
#include <hip/hip_runtime.h>
#include <math.h>

typedef __attribute__((ext_vector_type(2))) float v2f;
typedef __attribute__((ext_vector_type(8))) float v8f;

#define S_LEN 1024
#define KTILES (S_LEN / 16)   // 64 key tiles of 16

// Butterfly XOR shuffle via ds_swizzle_b32 (group-of-32 mode):
// offset = {1'b0, xor_mask[4:0], or_mask=0, and_mask=0x1F}
template <int MASK>
__device__ __forceinline__ float swz_xor(float v) {
    return __int_as_float(
        __builtin_amdgcn_ds_swizzle(__float_as_int(v), (MASK << 10) | 0x1F));
}

// One wave handles one 16-query tile against all 1024 keys (flash-attention,
// two-pass softmax). 8 waves per block -> 128 queries per block.
__global__ __launch_bounds__(256)
void SelfAttention_flash_wmma_kernel(const float* __restrict__ x,
                                     const float* __restrict__ Wq, const float* __restrict__ bq,
                                     const float* __restrict__ Wk, const float* __restrict__ bk,
                                     const float* __restrict__ Wv, const float* __restrict__ bv,
                                     float* __restrict__ out)
{
    __shared__ float k_lds[S_LEN][4];   // [key][dim0..2, pad0]  16 KB
    __shared__ float v_lds[S_LEN][4];   // [key][dim0..2, pad]   16 KB

    const int b    = blockIdx.x >> 3;          // batch index
    const int qb   = (blockIdx.x & 7) * 128;   // query base of this block
    const int tid  = threadIdx.x;
    const int lane = tid & 31;
    const int wave = tid >> 5;
    const int half = lane >> 4;                // 0: lanes 0-15, 1: lanes 16-31
    const int l16  = lane & 15;

    // ---- uniform weight loads ----
    float wq[9], wk[9], wv[9], bq3[3], bk3[3], bv3[3];
#pragma unroll
    for (int i = 0; i < 9; ++i) { wq[i] = Wq[i]; wk[i] = Wk[i]; wv[i] = Wv[i]; }
#pragma unroll
    for (int i = 0; i < 3; ++i) { bq3[i] = bq[i]; bk3[i] = bk[i]; bv3[i] = bv[i]; }

    // ---- cooperatively compute K,V for this batch row into LDS ----
    const float* xb = x + (size_t)b * S_LEN * 3;
    for (int i = tid; i < S_LEN; i += 256) {
        float x0 = xb[i * 3 + 0], x1 = xb[i * 3 + 1], x2 = xb[i * 3 + 2];
#pragma unroll
        for (int e = 0; e < 3; ++e) {
            k_lds[i][e] = wk[e * 3 + 0] * x0 + wk[e * 3 + 1] * x1 + wk[e * 3 + 2] * x2 + bk3[e];
            v_lds[i][e] = wv[e * 3 + 0] * x0 + wv[e * 3 + 1] * x1 + wv[e * 3 + 2] * x2 + bv3[e];
        }
        k_lds[i][3] = 0.0f;
        v_lds[i][3] = 0.0f;
    }
    __syncthreads();

    // ---- build Q tile in WMMA A-layout (16x4 f32, 2 VGPRs) ----
    // lanes 0-15 : V0 = q[m][0], V1 = q[m][1]   (m = l16)
    // lanes16-31 : V0 = q[m][2], V1 = q[m][3]=0 (K padded to 4)
    // Fold softmax scale AND log2(e) into q: scores come out in base-2 domain,
    // so pass 2 needs only a bare v_exp_f32.
    const int q0   = qb + wave * 16;
    const int qrow = q0 + l16;
    const float qscale = 0.57735026918962576f * 1.4426950408889634f; // log2(e)/sqrt(3)
    float xq0 = xb[qrow * 3 + 0], xq1 = xb[qrow * 3 + 1], xq2 = xb[qrow * 3 + 2];
    float c00 = half ? wq[6] : wq[0];
    float c01 = half ? wq[7] : wq[1];
    float c02 = half ? wq[8] : wq[2];
    float cb0 = half ? bq3[2] : bq3[0];
    v2f A;
    A.x = (c00 * xq0 + c01 * xq1 + c02 * xq2 + cb0) * qscale;
    A.y = half ? 0.0f
               : (wq[3] * xq0 + wq[4] * xq1 + wq[5] * xq2 + bq3[1]) * qscale;

    // ================= pass 1: row maxima (base-2 scores) =================
    // C/D layout: lane holds key col N = l16, VGPR r holds row M = r + 8*half
    float mx[8];
#pragma unroll
    for (int r = 0; r < 8; ++r) mx[r] = -3.0e38f;

    for (int kt = 0; kt < KTILES; ++kt) {
        const float2 kb = *(const float2*)&k_lds[kt * 16 + l16][half * 2];
        v2f Bm; Bm.x = kb.x; Bm.y = kb.y;
        v8f c = {};
        c = __builtin_amdgcn_wmma_f32_16x16x4_f32(false, A, false, Bm, (short)0, c,
                                                  false, false);
#pragma unroll
        for (int r = 0; r < 8; ++r) mx[r] = fmaxf(mx[r], c[r]);
    }
#pragma unroll
    for (int r = 0; r < 8; ++r) {
        mx[r] = fmaxf(mx[r], swz_xor<8>(mx[r]));
        mx[r] = fmaxf(mx[r], swz_xor<4>(mx[r]));
        mx[r] = fmaxf(mx[r], swz_xor<2>(mx[r]));
        mx[r] = fmaxf(mx[r], swz_xor<1>(mx[r]));
    }

    // ================= pass 2: exp2 + l + acc.v (per-lane partials) ========
    // Fold the max-subtraction into the WMMA accumulator: D = q*k + (-mx).
    v8f cinit;
#pragma unroll
    for (int r = 0; r < 8; ++r) cinit[r] = -mx[r];

    float lsum[8], acc0[8], acc1[8], acc2[8];
#pragma unroll
    for (int r = 0; r < 8; ++r) { lsum[r] = 0.f; acc0[r] = 0.f; acc1[r] = 0.f; acc2[r] = 0.f; }

    for (int kt = 0; kt < KTILES; ++kt) {
        const float2 kb = *(const float2*)&k_lds[kt * 16 + l16][half * 2];
        v2f Bm; Bm.x = kb.x; Bm.y = kb.y;
        v8f c = __builtin_amdgcn_wmma_f32_16x16x4_f32(false, A, false, Bm, (short)0,
                                                      cinit, false, false);
        const float4 vv = *(const float4*)&v_lds[kt * 16 + l16][0];
#pragma unroll
        for (int r = 0; r < 8; ++r) {
            float p = __builtin_amdgcn_exp2f(c[r]);   // bare v_exp_f32
            lsum[r] += p;
            acc0[r] += p * vv.x;
            acc1[r] += p * vv.y;
            acc2[r] += p * vv.z;
        }
    }

    // ---- butterfly add-reduce over the 16 lanes holding one row ----
#pragma unroll
    for (int r = 0; r < 8; ++r) {
        lsum[r] += swz_xor<8>(lsum[r]);  lsum[r] += swz_xor<4>(lsum[r]);
        lsum[r] += swz_xor<2>(lsum[r]);  lsum[r] += swz_xor<1>(lsum[r]);
        acc0[r] += swz_xor<8>(acc0[r]);  acc0[r] += swz_xor<4>(acc0[r]);
        acc0[r] += swz_xor<2>(acc0[r]);  acc0[r] += swz_xor<1>(acc0[r]);
        acc1[r] += swz_xor<8>(acc1[r]);  acc1[r] += swz_xor<4>(acc1[r]);
        acc1[r] += swz_xor<2>(acc1[r]);  acc1[r] += swz_xor<1>(acc1[r]);
        acc2[r] += swz_xor<8>(acc2[r]);  acc2[r] += swz_xor<4>(acc2[r]);
        acc2[r] += swz_xor<2>(acc2[r]);  acc2[r] += swz_xor<1>(acc2[r]);
    }

    // ---- store: lane (l16 == r) of each half writes row r + 8*half ----
    float* ob = out + ((size_t)b * S_LEN + q0) * 3;
#pragma unroll
    for (int r = 0; r < 8; ++r) {
        if (l16 == r) {
            const int m = r + half * 8;
            const float inv = 1.0f / lsum[r];
            ob[m * 3 + 0] = acc0[r] * inv;
            ob[m * 3 + 1] = acc1[r] * inv;
            ob[m * 3 + 2] = acc2[r] * inv;
        }
    }
}

extern "C" void kernel_launch(void* const* d_in, const int* in_sizes, int n_in,
                              void* d_out, int out_size, void* d_ws, size_t ws_size,
                              hipStream_t stream) {
    const float* x  = (const float*)d_in[0];
    const float* Wq = (const float*)d_in[1];
    const float* bq = (const float*)d_in[2];
    const float* Wk = (const float*)d_in[3];
    const float* bk = (const float*)d_in[4];
    const float* Wv = (const float*)d_in[5];
    const float* bv = (const float*)d_in[6];
    float* out = (float*)d_out;

    const int B = in_sizes[0] / (S_LEN * 3);   // 256
    dim3 grid(B * 8);                           // 8 query slabs of 128 per batch
    dim3 block(256);                            // 8 waves
    SelfAttention_flash_wmma_kernel<<<grid, block, 0, stream>>>(
        x, Wq, bq, Wk, bk, Wv, bv, out);
}